// MultiModalFusionGAT_25658134626985
// MI455X (gfx1250) — compile-verified
//
#include <hip/hip_runtime.h>
#include <math.h>

#define HDIM   512
#define NHEADS 8
#define DHEAD  64
#define NNODES 32768
#define NEDGES 262144

typedef float v2f __attribute__((ext_vector_type(2)));
typedef float v8f __attribute__((ext_vector_type(8)));

__device__ __forceinline__ float gelu_f(float x) {
    return 0.5f * x * (1.0f + erff(x * 0.70710678118654752f));
}

// order-preserving mapping float -> unsigned for atomicMax-based segment max
__device__ __forceinline__ unsigned fkey(float f) {
    unsigned u = __float_as_uint(f);
    return (u & 0x80000000u) ? ~u : (u | 0x80000000u);
}
__device__ __forceinline__ float funkey(unsigned k) {
    unsigned u = (k & 0x80000000u) ? (k ^ 0x80000000u) : ~k;
    return __uint_as_float(u);
}

// low 32 bits of a flat pointer into LDS == wave-relative LDS byte address
__device__ __forceinline__ unsigned lds_addr_of(const void* p) {
    return (unsigned)(uintptr_t)p;
}

// ---------------------------------------------------------------------------
// WMMA fp32 GEMM:  C[r, ccol] = sum_k A[r, acol + k] * B[k, bcol] (+bias) (+epilogue)
// Block computes a 128x64 output tile. 8 waves, each wave does 16 rows x 64 cols
// with 4 accumulators of v_wmma_f32_16x16x4_f32, K staged through LDS (chunks of 32).
// Tile staging uses gfx1250 GLOBAL_LOAD_ASYNC_TO_LDS_B128 (ASYNCcnt) when no
// input transform is needed; the gelu-input variant stages through VGPRs.
// rel_mode=1: per-head 64x64 relation GEMM, head = blockIdx.z
// resid!=null: out = beta*(1-R)*o + ((1-beta)*(1-R)+R)*resid, beta=sigmoid(*skip)
// ---------------------------------------------------------------------------
__global__ __launch_bounds__(256)
void wmma_gemm_kernel(const float* __restrict__ A, int lda, int a_col,
                      const float* __restrict__ B, int ldb,
                      const float* __restrict__ bias,
                      float* __restrict__ C, int ldc, int c_col,
                      int K, int rel_mode, int gelu_in,
                      const float* __restrict__ resid,
                      const float* __restrict__ skip_logit)
{
    __shared__ __align__(16) float As[128 * 36];   // 128 rows, 32 cols, pad to 36
    __shared__ __align__(16) float Bs[32 * 64];    // 32 rows (K), 64 cols (N)

    const int row0  = blockIdx.x * 128;
    const int head  = rel_mode ? blockIdx.z : 0;
    const int acol0 = a_col + head * 64;
    const float* Bp = B + (rel_mode ? (size_t)head * (size_t)K * (size_t)ldb : 0);
    const int bcol0 = blockIdx.y * 64;
    const int ccol0 = c_col + head * 64 + bcol0;

    const int tid  = threadIdx.x;
    const int lane = tid & 31;
    const int wave = tid >> 5;
    const int half = lane >> 4;     // 0: lanes 0-15, 1: lanes 16-31
    const int l15  = lane & 15;

    v8f acc[4] = {};

    for (int k0 = 0; k0 < K; k0 += 32) {
        if (!gelu_in) {
            // ---- async DMA staging: memory -> LDS, 16B per lane per op ----
            #pragma unroll
            for (int i = 0; i < 4; i++) {
                int idx = tid + i * 256;                 // 0..1023  (A: 128x32)
                int r   = idx >> 3;
                int cg  = (idx & 7) * 4;
                const void* gp = (const void*)(A + (size_t)(row0 + r) * lda + acol0 + k0 + cg);
                unsigned la = lds_addr_of(&As[r * 36 + cg]);
                asm volatile("global_load_async_to_lds_b128 %0, %1, off"
                             :: "v"(la), "v"(gp) : "memory");
            }
            #pragma unroll
            for (int i = 0; i < 2; i++) {
                int idx = tid + i * 256;                 // 0..511   (B: 32x64)
                int r   = idx >> 4;
                int cg  = (idx & 15) * 4;
                const void* gp = (const void*)(Bp + (size_t)(k0 + r) * ldb + bcol0 + cg);
                unsigned la = lds_addr_of(&Bs[r * 64 + cg]);
                asm volatile("global_load_async_to_lds_b128 %0, %1, off"
                             :: "v"(la), "v"(gp) : "memory");
            }
            asm volatile("s_wait_asynccnt 0" ::: "memory");
        } else {
            // ---- manual staging with GeLU applied to A elements ----
            #pragma unroll
            for (int i = 0; i < 4; i++) {
                int idx = tid + i * 256;
                int r   = idx >> 3;
                int cg  = (idx & 7) * 4;
                float4 v = *(const float4*)(A + (size_t)(row0 + r) * lda + acol0 + k0 + cg);
                v.x = gelu_f(v.x); v.y = gelu_f(v.y);
                v.z = gelu_f(v.z); v.w = gelu_f(v.w);
                *(float4*)&As[r * 36 + cg] = v;
            }
            #pragma unroll
            for (int i = 0; i < 2; i++) {
                int idx = tid + i * 256;
                int r   = idx >> 4;
                int cg  = (idx & 15) * 4;
                float4 v = *(const float4*)(Bp + (size_t)(k0 + r) * ldb + bcol0 + cg);
                *(float4*)&Bs[r * 64 + cg] = v;
            }
        }
        __syncthreads();

        const int arow = wave * 16 + l15;
        #pragma unroll
        for (int kk = 0; kk < 32; kk += 4) {
            // A fragment: lanes 0-15 -> K={kk,kk+1}, lanes 16-31 -> K={kk+2,kk+3}
            v2f a;
            a.x = As[arow * 36 + kk + half * 2];
            a.y = As[arow * 36 + kk + half * 2 + 1];
            #pragma unroll
            for (int n = 0; n < 4; n++) {
                int bc = n * 16 + l15;
                v2f b;
                b.x = Bs[(kk + half * 2) * 64 + bc];
                b.y = Bs[(kk + half * 2 + 1) * 64 + bc];
                acc[n] = __builtin_amdgcn_wmma_f32_16x16x4_f32(
                    false, a, false, b, (short)0, acc[n], false, false);
            }
        }
        __syncthreads();
    }

    float c1 = 1.0f, c2 = 0.0f;
    if (resid) {
        float beta = 1.0f / (1.0f + expf(-skip_logit[0]));
        c1 = beta * 0.7f;                   // beta*(1-R_ALPHA)
        c2 = (1.0f - beta) * 0.7f + 0.3f;   // (1-beta)*(1-R)+R
    }

    #pragma unroll
    for (int n = 0; n < 4; n++) {
        int ccol = ccol0 + n * 16 + l15;
        float bv = bias ? bias[ccol] : 0.0f;
        #pragma unroll
        for (int j = 0; j < 8; j++) {
            // C/D layout: VGPR j, lanes 0-15 -> M=j, lanes 16-31 -> M=j+8
            int r = row0 + wave * 16 + half * 8 + j;
            float o = acc[n][j] + bv;
            float out = resid ? (c1 * o + c2 * resid[(size_t)r * ldc + ccol]) : o;
            C[(size_t)r * ldc + ccol] = out;
        }
    }
}

// ---------------------------------------------------------------------------
// Edge pass 1: one wave per edge. logits[e,h] = <q[dst,h,:], krel[src,h,:]> * p_rel[h]/8
// Also tracks per-(dst,head) running max via order-preserving unsigned atomicMax.
// Element layout: chunk c, lane l covers elements c*128 + l*4 .. +3 -> head 2c + (l>=16)
// ---------------------------------------------------------------------------
__global__ __launch_bounds__(256)
void edge_logits_kernel(const float* __restrict__ q, const float* __restrict__ krel,
                        const int* __restrict__ src, const int* __restrict__ dst,
                        const float* __restrict__ p_rel,
                        float* __restrict__ logits, unsigned* __restrict__ segmax)
{
    int e    = blockIdx.x * 8 + (threadIdx.x >> 5);
    int lane = threadIdx.x & 31;
    int s = src[e], d = dst[e];
    const float4* qp = (const float4*)(q    + (size_t)d * HDIM);
    const float4* kp = (const float4*)(krel + (size_t)s * HDIM);
    #pragma unroll
    for (int c = 0; c < 4; c++) {
        float4 qv = qp[c * 32 + lane];
        float4 kv = kp[c * 32 + lane];
        float p = qv.x * kv.x + qv.y * kv.y + qv.z * kv.z + qv.w * kv.w;
        #pragma unroll
        for (int off = 8; off >= 1; off >>= 1) p += __shfl_xor(p, off, 32);
        int h = 2 * c + (lane >> 4);
        if ((lane & 15) == 0) {
            float lg = p * p_rel[h] * 0.125f;   // 1/sqrt(64)
            logits[(size_t)e * NHEADS + h] = lg;
            atomicMax(&segmax[(size_t)d * NHEADS + h], fkey(lg));
        }
    }
}

// Edge pass 2: per (edge, head): e = exp(logit - segmax[dst]); segsum[dst] += e
__global__ __launch_bounds__(256)
void edge_expsum_kernel(const int* __restrict__ dst, float* __restrict__ logits,
                        const unsigned* __restrict__ segmax, float* __restrict__ segsum)
{
    int t = blockIdx.x * 256 + threadIdx.x;   // t < NEDGES*NHEADS
    int e = t >> 3, h = t & 7;
    int d = dst[e];
    float m  = funkey(segmax[(size_t)d * NHEADS + h]);
    float ev = expf(logits[t] - m);
    logits[t] = ev;
    atomicAdd(&segsum[(size_t)d * NHEADS + h], ev);
}

// Edge pass 3: one wave per edge: agg[dst,h,:] += alpha[e,h] * vrel[src,h,:]
__global__ __launch_bounds__(256)
void edge_aggregate_kernel(const float* __restrict__ vrel, const float* __restrict__ expv,
                           const float* __restrict__ segsum,
                           const int* __restrict__ src, const int* __restrict__ dst,
                           float* __restrict__ agg)
{
    int e    = blockIdx.x * 8 + (threadIdx.x >> 5);
    int lane = threadIdx.x & 31;
    int s = src[e], d = dst[e];
    const float4* vp = (const float4*)(vrel + (size_t)s * HDIM);
    float* ap = agg + (size_t)d * HDIM;
    #pragma unroll
    for (int c = 0; c < 4; c++) {
        int h = 2 * c + (lane >> 4);
        float alpha = expv[(size_t)e * NHEADS + h] /
                      (segsum[(size_t)d * NHEADS + h] + 1e-16f);
        float4 vv = vp[c * 32 + lane];
        int base = c * 128 + lane * 4;
        atomicAdd(ap + base + 0, alpha * vv.x);
        atomicAdd(ap + base + 1, alpha * vv.y);
        atomicAdd(ap + base + 2, alpha * vv.z);
        atomicAdd(ap + base + 3, alpha * vv.w);
    }
}

// ---------------------------------------------------------------------------
extern "C" void kernel_launch(void* const* d_in, const int* in_sizes, int n_in,
                              void* d_out, int out_size, void* d_ws, size_t ws_size,
                              hipStream_t stream)
{
    const float* x_user  = (const float*)d_in[0];
    const float* x_item  = (const float*)d_in[1];
    const int*   edge_ui = (const int*)d_in[2];
    const int*   edge_iu = (const int*)d_in[3];
    int pi = 4;
    // params in insertion order: per node type {k,q,v,o} x {w,b}, then skip
    const float* k_w_user = (const float*)d_in[pi++]; const float* k_b_user = (const float*)d_in[pi++];
    const float* q_w_user = (const float*)d_in[pi++]; const float* q_b_user = (const float*)d_in[pi++];
    const float* v_w_user = (const float*)d_in[pi++]; const float* v_b_user = (const float*)d_in[pi++];
    const float* o_w_user = (const float*)d_in[pi++]; const float* o_b_user = (const float*)d_in[pi++];
    const float* skip_user = (const float*)d_in[pi++];
    const float* k_w_item = (const float*)d_in[pi++]; const float* k_b_item = (const float*)d_in[pi++];
    const float* q_w_item = (const float*)d_in[pi++]; const float* q_b_item = (const float*)d_in[pi++];
    const float* v_w_item = (const float*)d_in[pi++]; const float* v_b_item = (const float*)d_in[pi++];
    const float* o_w_item = (const float*)d_in[pi++]; const float* o_b_item = (const float*)d_in[pi++];
    const float* skip_item = (const float*)d_in[pi++];
    const float* a_rel_ui = (const float*)d_in[pi++]; const float* m_rel_ui = (const float*)d_in[pi++];
    const float* p_rel_ui = (const float*)d_in[pi++];
    const float* a_rel_iu = (const float*)d_in[pi++]; const float* m_rel_iu = (const float*)d_in[pi++];
    const float* p_rel_iu = (const float*)d_in[pi++];

    // workspace layout (floats)
    float* ws = (float*)d_ws;
    const size_t NB = (size_t)NNODES * HDIM;               // 16.78M floats per node buffer
    float*    tmp      = ws;                               // raw k/v projection
    float*    qbuf     = ws + 1 * NB;                      // q of destination type
    float*    krel     = ws + 2 * NB;
    float*    vrel     = ws + 3 * NB;
    float*    agg_item = ws + 4 * NB;
    float*    agg_user = ws + 5 * NB;
    float*    logits   = ws + 6 * NB;                      // NEDGES*NHEADS
    unsigned* segmax   = (unsigned*)(ws + 6 * NB + (size_t)NEDGES * NHEADS);
    float*    segsum   = ws + 6 * NB + (size_t)NEDGES * NHEADS + (size_t)NNODES * NHEADS;

    auto gemm = [&](const float* A, const float* B, const float* bias, float* C,
                    int gelu, const float* resid, const float* skip) {
        dim3 grid(NNODES / 128, HDIM / 64, 1);
        wmma_gemm_kernel<<<grid, 256, 0, stream>>>(A, HDIM, 0, B, HDIM, bias,
                                                   C, HDIM, 0, HDIM, 0, gelu, resid, skip);
    };
    auto relgemm = [&](const float* A, const float* R, float* C) {
        dim3 grid(NNODES / 128, 1, NHEADS);
        wmma_gemm_kernel<<<grid, 256, 0, stream>>>(A, HDIM, 0, R, DHEAD, nullptr,
                                                   C, HDIM, 0, DHEAD, 1, 0, nullptr, nullptr);
    };
    auto edge_phase = [&](const float* q, const float* kr, const float* vr,
                          const int* eidx, const float* p_rel, float* agg) {
        hipMemsetAsync(segmax, 0, (size_t)NNODES * NHEADS * sizeof(unsigned), stream);
        hipMemsetAsync(segsum, 0, (size_t)NNODES * NHEADS * sizeof(float), stream);
        hipMemsetAsync(agg,    0, NB * sizeof(float), stream);
        const int* src = eidx;
        const int* dst = eidx + NEDGES;
        edge_logits_kernel<<<NEDGES / 8, 256, 0, stream>>>(q, kr, src, dst, p_rel, logits, segmax);
        edge_expsum_kernel<<<(NEDGES * NHEADS) / 256, 256, 0, stream>>>(dst, logits, segmax, segsum);
        edge_aggregate_kernel<<<NEDGES / 8, 256, 0, stream>>>(vr, logits, segsum, src, dst, agg);
    };

    // ---- edge type ui: src=user, dst=item ----
    gemm(x_item, q_w_item, q_b_item, qbuf, 0, nullptr, nullptr);
    gemm(x_user, k_w_user, k_b_user, tmp, 0, nullptr, nullptr);
    relgemm(tmp, a_rel_ui, krel);
    gemm(x_user, v_w_user, v_b_user, tmp, 0, nullptr, nullptr);
    relgemm(tmp, m_rel_ui, vrel);
    edge_phase(qbuf, krel, vrel, edge_ui, p_rel_ui, agg_item);

    // ---- edge type iu: src=item, dst=user ----
    gemm(x_user, q_w_user, q_b_user, qbuf, 0, nullptr, nullptr);
    gemm(x_item, k_w_item, k_b_item, tmp, 0, nullptr, nullptr);
    relgemm(tmp, a_rel_iu, krel);
    gemm(x_item, v_w_item, v_b_item, tmp, 0, nullptr, nullptr);
    relgemm(tmp, m_rel_iu, vrel);
    edge_phase(qbuf, krel, vrel, edge_iu, p_rel_iu, agg_user);

    // ---- node outputs: gelu(agg) @ o_w + o_b, skip gate + residual blend ----
    float* out_user = (float*)d_out;
    float* out_item = out_user + NB;
    gemm(agg_user, o_w_user, o_b_user, out_user, 1, x_user, skip_user);
    gemm(agg_item, o_w_item, o_b_item, out_item, 1, x_item, skip_item);
}